// WeightedPool1D_74569222193663
// MI455X (gfx1250) — compile-verified
//
#include <hip/hip_runtime.h>
#include <stdint.h>

// Problem constants (from reference): B=8, N=100000, K=1024, D=128.
#define BB    8
#define NN    100000
#define KK    1024
#define DD    128
#define DTILE 32
#define NDT   (DD / DTILE)   // 4 d-tiles
#define BLOCK 256
#define SGN   32             // nodes staged per pipeline stage (4 KB)
#define PDEPTH 4             // async pipeline depth (ring of 4 stages)

// ---- CDNA5 feature probes -------------------------------------------------
#if defined(__has_builtin)
#  if __has_builtin(__builtin_amdgcn_global_load_async_to_lds_b128)
#    define HAVE_ASYNC_LDS 1
#  endif
#  if __has_builtin(__builtin_amdgcn_s_wait_asynccnt)
#    define HAVE_WAIT_ASYNC 1
#  endif
#endif

#if defined(HAVE_WAIT_ASYNC)
#  define WAIT_ASYNC(n) __builtin_amdgcn_s_wait_asynccnt(n)
#elif defined(HAVE_ASYNC_LDS)
#  define WAIT_ASYNC(n) asm volatile("s_wait_asynccnt %0" :: "i"(n) : "memory")
#else
#  define WAIT_ASYNC(n)
#endif

// Builtin prototype (confirmed by hipcc diagnostic): (v4i AS1*, v4i AS3*,
// imm offset, imm cpol).
typedef int v4i __attribute__((vector_size(4 * sizeof(int))));
typedef __attribute__((address_space(1))) v4i as1_v4i;   // global
typedef __attribute__((address_space(3))) v4i as3_v4i;   // LDS

// ---- phase 0: compact per-node weights wc[i] = weight[seg[i]*N + i] -------
__global__ void __launch_bounds__(BLOCK)
phase0_gather(const int* __restrict__ seg, const float* __restrict__ weight,
              float* __restrict__ wc) {
  int i = blockIdx.x * BLOCK + threadIdx.x;
  if (i < NN) wc[i] = weight[(size_t)seg[i] * NN + i];
}

// ---- phase 1: per-(chunk, d-tile, batch) LDS segment accumulation ---------
__global__ void __launch_bounds__(BLOCK)
phase1_segsum(const float* __restrict__ x, const int* __restrict__ seg,
              const float* __restrict__ wc, float* __restrict__ partial,
              int nodes_per_chunk, int nchunks) {
  __shared__ float acc[KK * DTILE];                // 128 KB accumulator
  __shared__ float xbuf[PDEPTH * SGN * DTILE];     // 16 KB staging ring

  const int tid   = threadIdx.x;
  const int lane  = tid & 31;
  const int wv    = tid >> 5;                      // 0..7 (wave32)
  const int chunk = blockIdx.x;
  const int dt    = blockIdx.y;
  const int b     = blockIdx.z;
  const int d0    = dt * DTILE;

  // zero the accumulator
  float4* acc4 = (float4*)acc;
  for (int i = tid; i < KK * DTILE / 4; i += BLOCK)
    acc4[i] = make_float4(0.f, 0.f, 0.f, 0.f);
  __syncthreads();

  const long chunk_beg = (long)chunk * nodes_per_chunk;
  const long chunk_end = (chunk_beg + nodes_per_chunk < (long)NN)
                             ? chunk_beg + nodes_per_chunk : (long)NN;
  const int nstages = (int)((chunk_end - chunk_beg + SGN - 1) / SGN);

  // Lane -> (row within this wave's 4 rows, 16B column). Wave wv stages and
  // consumes rows 4wv..4wv+3 of each 32-node stage: per-wave ASYNCcnt is a
  // sufficient dependency fence; no cross-wave barriers in the stream loop.
  const int rr    = lane >> 3;                     // 0..3
  const int cc    = lane & 7;                      // 0..7 (8 x 16B = 128B row)
  const int myrow = wv * 4 + rr;                   // 0..31

  const float* xbase = x + (size_t)b * NN * DD + d0 + cc * 4;

  auto issue = [&](int s) {
#if defined(HAVE_ASYNC_LDS)
    long node   = chunk_beg + (long)s * SGN + myrow;
    long nclamp = node < (long)NN ? node : (long)(NN - 1);
    const float* g = xbase + nclamp * (long)DD;
    float*       l = &xbuf[((s & (PDEPTH - 1)) * SGN + myrow) * DTILE + cc * 4];
    __builtin_amdgcn_global_load_async_to_lds_b128((as1_v4i*)g, (as3_v4i*)l,
                                                   0, 0);
#else
    (void)s;
#endif
  };

#if defined(HAVE_ASYNC_LDS)
  for (int s = 0; s < PDEPTH && s < nstages; ++s) issue(s);
#endif

  for (int s = 0; s < nstages; ++s) {
#if defined(HAVE_ASYNC_LDS)
    // Async loads complete in order per wave. Outstanding after stage s
    // completes: stages (s+1 .. min(nstages, s+PDEPTH)-1).
    int rem = nstages - 1 - s;
    if (rem > PDEPTH - 1) rem = PDEPTH - 1;
    switch (rem) {                   // immediates must be literals
      case 0: WAIT_ASYNC(0); break;
      case 1: WAIT_ASYNC(1); break;
      case 2: WAIT_ASYNC(2); break;
      default: WAIT_ASYNC(3); break;
    }
    const float* xb = &xbuf[(s & (PDEPTH - 1)) * SGN * DTILE];
#endif
    const long base = chunk_beg + (long)s * SGN;
    for (int r4 = 0; r4 < 4; ++r4) {
      long i = base + wv * 4 + r4;
      if (i >= chunk_end) break;
      int   sgi = seg[i];                          // wave-uniform broadcast
      float w   = wc[i];
#if defined(HAVE_ASYNC_LDS)
      float xv = xb[(wv * 4 + r4) * DTILE + lane];
#else
      float xv = x[((size_t)b * NN + i) * DD + d0 + lane];
#endif
      atomicAdd(&acc[(size_t)sgi * DTILE + lane], w * xv);  // ds_add_f32
    }
#if defined(HAVE_ASYNC_LDS)
    if (s + PDEPTH < nstages) issue(s + PDEPTH);
#endif
  }
#if defined(HAVE_ASYNC_LDS)
  WAIT_ASYNC(0);
#endif
  __syncthreads();

  // flush partial (plain streaming stores; no global atomics)
  float4* pp4 = (float4*)(partial +
      (((size_t)b * NDT + dt) * nchunks + chunk) * (size_t)(KK * DTILE));
  for (int i = tid; i < KK * DTILE / 4; i += BLOCK) pp4[i] = acc4[i];
}

// ---- phase 2: reduce chunk partials -> out (writes every element) ---------
__global__ void __launch_bounds__(BLOCK)
phase2_reduce(const float* __restrict__ partial, float* __restrict__ out,
              int nchunks) {
  int g = blockIdx.x * BLOCK + threadIdx.x;
  if (g >= BB * KK * DD) return;
  int l  = g & (DTILE - 1);
  int k  = (g / DTILE) & (KK - 1);
  int dt = (g / (DTILE * KK)) & (NDT - 1);
  int b  = g / (DTILE * KK * NDT);
  const float* p = partial +
      (((size_t)b * NDT + dt) * nchunks) * (size_t)(KK * DTILE) +
      (size_t)k * DTILE + l;
  float s = 0.f;
  for (int c = 0; c < nchunks; ++c) s += p[(size_t)c * (KK * DTILE)];
  out[((size_t)b * KK + k) * DD + dt * DTILE + l] = s;
}

// ---- host launcher --------------------------------------------------------
extern "C" void kernel_launch(void* const* d_in, const int* in_sizes, int n_in,
                              void* d_out, int out_size, void* d_ws,
                              size_t ws_size, hipStream_t stream) {
  const float* x      = (const float*)d_in[0];
  const float* weight = (const float*)d_in[1];
  const int*   seg    = (const int*)d_in[2];
  float*       out    = (float*)d_out;

  // workspace layout: [ wc: N floats | partials: C * B * 4 * K * 32 floats ]
  size_t wc_bytes = (((size_t)NN * sizeof(float)) + 255) & ~(size_t)255;
  int C = 16;                                    // node chunks (parallelism)
  while (C > 1 &&
         wc_bytes + (size_t)C * BB * NDT * KK * DTILE * sizeof(float) > ws_size)
    C >>= 1;
  float* wc      = (float*)d_ws;
  float* partial = (float*)((char*)d_ws + wc_bytes);
  int npc = (NN + C - 1) / C;

  phase0_gather<<<(NN + BLOCK - 1) / BLOCK, BLOCK, 0, stream>>>(seg, weight, wc);
  dim3 g1(C, NDT, BB);
  phase1_segsum<<<g1, BLOCK, 0, stream>>>(x, seg, wc, partial, npc, C);
  phase2_reduce<<<(BB * KK * DD + BLOCK - 1) / BLOCK, BLOCK, 0, stream>>>(
      partial, out, C);
}